// KblamGemma3nAttention_29884382446227
// MI455X (gfx1250) — compile-verified
//
#include <hip/hip_runtime.h>
#include <hip/hip_bf16.h>
#include <math.h>

// ---------------- problem constants (match reference) ----------------
#define B_   2
#define S_   1024
#define H_   2048
#define NH_  16
#define NKV_ 4
#define HD_  128
#define KB_  1024
#define NREP_ (NH_ / NKV_)
#define SCALE_ 0.08838834764831845f   // 1/sqrt(128)
#define THETA_ 10000.0f

typedef float v2f __attribute__((ext_vector_type(2)));
typedef float v8f __attribute__((ext_vector_type(8)));

// D = A(16x4 f32) * B(4x16 f32) + C(16x16 f32), full fp32 matrix pipe on gfx1250
__device__ __forceinline__ v8f wmma_f32_k4(v2f a, v2f b, v8f c) {
    return __builtin_amdgcn_wmma_f32_16x16x4_f32(
        /*neg_a=*/false, a, /*neg_b=*/false, b,
        /*c_mod=*/(short)0, c, /*reuse_a=*/false, /*reuse_b=*/false);
}

// =====================================================================
// Register-blocked NT GEMM: Y[M,N] = X[M,K] * W[N,K]^T (fp32, row-major)
// One 64x64 output tile per wave (4 A-frags x 4 B-frags = 16 v8f acc):
// 16 operand loads per 16 WMMAs -> 4x less load traffic per FLOP than
// the naive 16x16/wave version. 8 waves per block.
// =====================================================================
__global__ __launch_bounds__(256) void gemm_nt_kernel(
    const float* __restrict__ X, const float* __restrict__ W,
    float* __restrict__ Y, int M, int N, int K)
{
    const int wave = threadIdx.x >> 5;
    const int lane = threadIdx.x & 31;
    const int ntn  = N >> 6;                      // 64-wide tiles
    const int ntm  = M >> 6;
    int tile = blockIdx.x * (blockDim.x >> 5) + wave;
    if (tile >= ntm * ntn) return;                // wave-uniform: EXEC stays full
    const int tm = tile / ntn;
    const int tn = tile % ntn;

    const int mr   = lane & 15;                   // A-frag row / B-frag col
    const int koff = (lane >> 4) * 2;             // K sub-offset per lane half

    const float* xr[4];
    const float* wr[4];
    #pragma unroll
    for (int i = 0; i < 4; ++i) {
        xr[i] = X + (size_t)(tm * 64 + 16 * i + mr) * K;
        wr[i] = W + (size_t)(tn * 64 + 16 * i + mr) * K;
    }

    v8f acc[4][4];
    #pragma unroll
    for (int i = 0; i < 4; ++i)
        #pragma unroll
        for (int j = 0; j < 4; ++j) { v8f z = {}; acc[i][j] = z; }

    for (int k = 0; k < K; k += 4) {
        v2f a[4], b[4];
        #pragma unroll
        for (int i = 0; i < 4; ++i) {
            a[i].x = xr[i][k + koff];  a[i].y = xr[i][k + koff + 1];
            b[i].x = wr[i][k + koff];  b[i].y = wr[i][k + koff + 1];
        }
        #pragma unroll
        for (int i = 0; i < 4; ++i)
            #pragma unroll
            for (int j = 0; j < 4; ++j)
                acc[i][j] = wmma_f32_k4(a[i], b[j], acc[i][j]);
    }

    // C/D layout: VGPR r -> row (r + 8*laneHalf), col = lane&15
    #pragma unroll
    for (int i = 0; i < 4; ++i) {
        const int rbase = tm * 64 + 16 * i + 8 * (lane >> 4);
        #pragma unroll
        for (int j = 0; j < 4; ++j) {
            const int col = tn * 64 + 16 * j + mr;
            #pragma unroll
            for (int r = 0; r < 8; ++r)
                Y[(size_t)(rbase + r) * N + col] = acc[i][j][r];
        }
    }
}

// =====================================================================
// RoPE, in place on (B, S, heads, HD) fp32. One thread per (b,s,h,pair).
// =====================================================================
__global__ void rope_kernel(float* __restrict__ q,
                            const int* __restrict__ pos_ids, int heads)
{
    size_t idx   = (size_t)blockIdx.x * blockDim.x + threadIdx.x;
    size_t total = (size_t)B_ * S_ * heads * (HD_ / 2);
    if (idx >= total) return;
    int i = (int)(idx & 63); size_t t = idx >> 6;
    int h = (int)(t % heads); t /= heads;
    int s = (int)(t % S_);    int b = (int)(t / S_);

    float pos  = (float)pos_ids[b * S_ + s];
    float freq = __expf(-((float)(2 * i) / (float)HD_) * __logf(THETA_));
    float ang  = pos * freq;
    float c = __cosf(ang), sn = __sinf(ang);

    size_t base = (((size_t)b * S_ + s) * heads + h) * HD_;
    float x1 = q[base + i], x2 = q[base + i + 64];
    q[base + i]      = x1 * c - x2 * sn;
    q[base + i + 64] = x2 * c + x1 * sn;
}

// =====================================================================
// Flash attention over [KB keys (unmasked) ; S self keys (causal)].
// 4 waves/WG; each wave = one 16-query tile. K/V tiles staged through
// LDS with coalesced b128 loads; rows padded (132 / 17 floats) to keep
// fragment reads bank-conflict-free. All matmuls via fp32 WMMA.
// =====================================================================
__global__ __launch_bounds__(128) void attn_kernel(
    const float* __restrict__ qnew,  // (B,S,NH*HD)  not roped
    const float* __restrict__ qrop,  // (B,S,NH*HD)  roped
    const float* __restrict__ kself, // (B,S,NKV*HD) roped
    const float* __restrict__ vself, // (B,S,NKV*HD)
    const float* __restrict__ kbk,   // (B,KB,NKV*HD)
    const float* __restrict__ kbv,   // (B,KB,NKV*HD)
    float* __restrict__ outh)        // (B,S,NH*HD)
{
    __shared__ __align__(16) float kvbuf[4][16 * 132]; // per-wave K/V tile (padded)
    __shared__ float pbuf[4][16 * 17];                 // per-wave P tile (padded)

    const int wave = threadIdx.x >> 5;
    const int lane = threadIdx.x & 31;
    int wg = blockIdx.x;                          // ((b*NH + h)*(S/64) + qt)
    const int qt = wg % (S_ / 64); wg /= (S_ / 64);
    const int h  = wg % NH_;
    const int b  = wg / NH_;
    const int kv = h / NREP_;

    const int q0   = qt * 64 + wave * 16;         // first query row of this wave
    const int mr   = lane & 15;                   // A row / B col within tile
    const int koff = (lane >> 4) * 2;
    const int hl   = lane >> 4;
    const int nn   = lane & 15;

    float* kw = kvbuf[wave];
    float* pb = pbuf[wave];

    float m_run[8], l_run[8];
    v8f acc[8];
    #pragma unroll
    for (int r = 0; r < 8; ++r) { m_run[r] = -1e30f; l_run[r] = 0.0f; }
    #pragma unroll
    for (int t = 0; t < 8; ++t) { v8f z = {}; acc[t] = z; }

    float areg[64];                               // lane's half of the 16x128 Q tile

    for (int phase = 0; phase < 2; ++phase) {
        const float* Q     = (phase == 0) ? qnew : qrop;
        const float* Kp    = (phase == 0) ? kbk  : kself;
        const float* Vp    = (phase == 0) ? kbv  : vself;
        const int    nkeys = (phase == 0) ? KB_  : S_;

        // load A fragments for this phase's Q
        const float* qrow = Q + (((size_t)b * S_ + (q0 + mr)) * NH_ + h) * HD_;
        #pragma unroll
        for (int c = 0; c < 32; ++c) {
            areg[2 * c]     = qrow[4 * c + koff];
            areg[2 * c + 1] = qrow[4 * c + koff + 1];
        }

        for (int kt = 0; kt < nkeys / 16; ++kt) {
            const int kbase = kt * 16;
            if (phase == 1 && kbase > q0 + 15) break;   // causal: wave-uniform skip

            // ---- stage first 8 keys x 32 dims of K tile into LDS, coalesced ----
            {
                const float* kb0 =
                    Kp + (((size_t)b * nkeys + kbase) * NKV_ + kv) * HD_;
                #pragma unroll
                for (int hf = 0; hf < 2; ++hf) {
                    int c4  = lane + 32 * hf;           // float4 index 0..63
                    int row = c4 >> 3, col = (c4 & 7) * 4;
                    float4 t = *(const float4*)(kb0 + (size_t)row * (NKV_ * HD_) + col);
                    *(float4*)&kw[row * 132 + col] = t;
                }
            }
            asm volatile("s_wait_dscnt 0" ::: "memory");   // intra-wave DS visibility

            // ---- scores: S = Q * K^T (K-loop over HD in chunks of 4) ----
            v8f s = {};
            #pragma unroll
            for (int c = 0; c < 32; ++c) {
                v2f a, bb;
                a.x  = areg[2 * c];                a.y  = areg[2 * c + 1];
                bb.x = kw[nn * 132 + 4 * c + koff];
                bb.y = kw[nn * 132 + 4 * c + koff + 1];
                s = wmma_f32_k4(a, bb, s);
            }

            // ---- streaming softmax update ----
            float pv[8], f[8];
            #pragma unroll
            for (int r = 0; r < 8; ++r) {
                float sc = s[r] * SCALE_;
                if (phase == 1) {
                    int qi = q0 + 8 * hl + r;
                    int kj = kbase + nn;
                    if (kj > qi) sc = -1e30f;          // causal mask
                }
                float tm = sc;                         // row-max over 16 cols
                tm = fmaxf(tm, __shfl_xor(tm, 1, 32));
                tm = fmaxf(tm, __shfl_xor(tm, 2, 32));
                tm = fmaxf(tm, __shfl_xor(tm, 4, 32));
                tm = fmaxf(tm, __shfl_xor(tm, 8, 32));
                float mnew = fmaxf(m_run[r], tm);
                float p    = __expf(sc - mnew);
                f[r]       = __expf(m_run[r] - mnew);
                m_run[r]   = mnew;
                float ts = p;                          // row-sum over 16 cols
                ts += __shfl_xor(ts, 1, 32);
                ts += __shfl_xor(ts, 2, 32);
                ts += __shfl_xor(ts, 4, 32);
                ts += __shfl_xor(ts, 8, 32);
                l_run[r] = l_run[r] * f[r] + ts;
                pv[r] = p;
            }
            #pragma unroll
            for (int t = 0; t < 8; ++t)
                #pragma unroll
                for (int r = 0; r < 8; ++r)
                    acc[t][r] *= f[r];

            // ---- write P tile + stage V tile (reuses kw; DS is in-order) ----
            #pragma unroll
            for (int r = 0; r < 8; ++r)
                pb[(8 * hl + r) * 17 + nn] = pv[r];
            {
                const float* vb0 =
                    Vp + (((size_t)b * nkeys + kbase) * NKV_ + kv) * HD_;
                #pragma unroll
                for (int hf = 0; hf < 2; ++hf) {
                    int c4  = lane + 32 * hf;
                    int row = c4 >> 3, col = (c4 & 7) * 4;
                    float4 t = *(const float4*)(vb0 + (size_t)row * (NKV_ * HD_) + col);
                    *(float4*)&kw[row * 132 + col] = t;
                }
            }
            asm volatile("s_wait_dscnt 0" ::: "memory");

            // ---- O += P * V ----
            #pragma unroll
            for (int c2 = 0; c2 < 4; ++c2) {
                v2f pa;
                pa.x = pb[mr * 17 + 4 * c2 + koff];
                pa.y = pb[mr * 17 + 4 * c2 + koff + 1];
                const int key0 = 4 * c2 + koff;
                #pragma unroll
                for (int t = 0; t < 8; ++t) {
                    v2f vbf;
                    vbf.x = kw[key0 * 132 + 16 * t + nn];
                    vbf.y = kw[(key0 + 1) * 132 + 16 * t + nn];
                    acc[t] = wmma_f32_k4(pa, vbf, acc[t]);
                }
            }
        }
    }

    // ---- normalize + store per-head output ----
    const int rowg = q0 + 8 * hl;
    #pragma unroll
    for (int t = 0; t < 8; ++t) {
        #pragma unroll
        for (int r = 0; r < 8; ++r) {
            float o = acc[t][r] / l_run[r];
            outh[(((size_t)b * S_ + (rowg + r)) * NH_ + h) * HD_ + 16 * t + nn] = o;
        }
    }
}

// =====================================================================
// Launch
// =====================================================================
extern "C" void kernel_launch(void* const* d_in, const int* in_sizes, int n_in,
                              void* d_out, int out_size, void* d_ws, size_t ws_size,
                              hipStream_t stream) {
    const float* x    = (const float*)d_in[0];   // hidden_states (B,S,H)
    /* d_in[1] attention_mask: causal, applied analytically */
    const int*   pos  = (const int*)  d_in[2];   // position_ids (B,S)
    const float* kbk  = (const float*)d_in[3];   // kb_keys   (B,KB,NKV*HD)
    const float* kbv  = (const float*)d_in[4];   // kb_values (B,KB,NKV*HD)
    const float* Wq   = (const float*)d_in[5];   // (NH*HD, H)
    const float* Wqn  = (const float*)d_in[6];   // (NH*HD, H)
    const float* Wk   = (const float*)d_in[7];   // (NKV*HD, H)
    const float* Wv   = (const float*)d_in[8];   // (NKV*HD, H)
    const float* Wo   = (const float*)d_in[9];   // (H, NH*HD)
    float* out = (float*)d_out;

    float* ws   = (float*)d_ws;
    float* q    = ws;                    //  4M floats (B,S,NH*HD)
    float* qn   = ws + 4194304;          //  4M floats
    float* k    = ws + 8388608;          //  1M floats (B,S,NKV*HD)
    float* v    = ws + 9437184;          //  1M floats
    float* outh = ws + 10485760;         //  4M floats (B,S,NH*HD)

    const int M = B_ * S_;               // 2048
    dim3 blk(256);

    // 64x64 tiles per wave, 8 waves per block
    gemm_nt_kernel<<<(M / 64) * (2048 / 64) / 8, blk, 0, stream>>>(x, Wq,  q,  M, 2048, H_);
    gemm_nt_kernel<<<(M / 64) * (2048 / 64) / 8, blk, 0, stream>>>(x, Wqn, qn, M, 2048, H_);
    gemm_nt_kernel<<<(M / 64) * (512  / 64) / 8, blk, 0, stream>>>(x, Wk,  k,  M, 512,  H_);
    gemm_nt_kernel<<<(M / 64) * (512  / 64) / 8, blk, 0, stream>>>(x, Wv,  v,  M, 512,  H_);

    rope_kernel<<<(B_ * S_ * NH_  * 64 + 255) / 256, 256, 0, stream>>>(q, pos, NH_);
    rope_kernel<<<(B_ * S_ * NKV_ * 64 + 255) / 256, 256, 0, stream>>>(k, pos, NKV_);

    attn_kernel<<<B_ * NH_ * (S_ / 64), 128, 0, stream>>>(qn, q, k, v, kbk, kbv, outh);

    gemm_nt_kernel<<<(M / 64) * (2048 / 64) / 8, blk, 0, stream>>>(outh, Wo, out, M, 2048, 2048);
}